// Head_43765716746377
// MI455X (gfx1250) — compile-verified
//
#include <hip/hip_runtime.h>

#define B_ 8
#define T_ 2048
#define C_ 1024
#define H_ 64

typedef float v2f __attribute__((ext_vector_type(2)));
typedef float v8f __attribute__((ext_vector_type(8)));

#define WMMA_F32(a, b, c) \
    __builtin_amdgcn_wmma_f32_16x16x4_f32(false, (a), false, (b), (short)0, (c), false, false)

// Async global->LDS copy, 16 bytes per lane.  Tracked by ASYNCcnt.
__device__ __forceinline__ void async_copy_b128(void* lds, const void* g) {
    const uint32_t l = (uint32_t)(uintptr_t)lds;   // low 32 bits = LDS offset
    asm volatile("global_load_async_to_lds_b128 %0, %1, off"
                 :: "v"(l), "v"(g) : "memory");
}
#define WAIT_ASYNC(n) asm volatile("s_wait_asynccnt " #n ::: "memory")

// ---------------------------------------------------------------------------
// Stage 1: QKV projection.  [B*T, C] x [C, H] for Wq, Wk, Wv.
// Grid: (B*T)/64 blocks, 384 threads (12 waves).
// Wave w: matrix = w/4 (q,k,v), ntile = w%4 (16-col strip), 4 row-tiles.
// x chunks double-buffered in LDS via async copies.
// ---------------------------------------------------------------------------
__global__ __launch_bounds__(384)
void qkv_proj_kernel(const float* __restrict__ x,
                     const float* __restrict__ Wq,
                     const float* __restrict__ Wk,
                     const float* __restrict__ Wv,
                     float* __restrict__ Q,
                     float* __restrict__ Ko,
                     float* __restrict__ Vo)
{
    __shared__ float xs[2][64][68];   // stride 68 -> 16B-aligned rows, conflict-free cols

    const int tid   = threadIdx.x;
    const int wave  = tid >> 5;
    const int lane  = tid & 31;
    const int lhalf = lane >> 4;   // 0 or 1
    const int lmod  = lane & 15;
    const int mat   = wave >> 2;   // 0=Q 1=K 2=V
    const int ntile = wave & 3;    // 16-col tile of H
    const long rowBase = (long)blockIdx.x * 64;

    const float* W  = (mat == 0) ? Wq : (mat == 1) ? Wk : Wv;
    float*      Out = (mat == 0) ? Q  : (mat == 1) ? Ko : Vo;

    // 64x64 floats = 1024 float4 per chunk; 384 threads -> 3 issues each
    // (tail clamped so every wave issues exactly 3 -> uniform ASYNCcnt).
    auto issue_x = [&](int kc, int buf) {
        #pragma unroll
        for (int it = 0; it < 3; ++it) {
            int e = tid + it * 384;
            e = (e < 1023) ? e : 1023;
            const int r = e >> 4, c = (e & 15) << 2;
            async_copy_b128(&xs[buf][r][c], &x[(rowBase + r) * C_ + kc * 64 + c]);
        }
    };

    v8f acc[4];
    #pragma unroll
    for (int r = 0; r < 4; ++r) acc[r] = (v8f){0,0,0,0,0,0,0,0};

    issue_x(0, 0);
    for (int kc = 0; kc < C_ / 64; ++kc) {
        const int cur = kc & 1;
        if (kc + 1 < C_ / 64) {
            issue_x(kc + 1, cur ^ 1);
            WAIT_ASYNC(3);          // 3 next-chunk copies may stay in flight
        } else {
            WAIT_ASYNC(0);
        }
        __syncthreads();

        #pragma unroll
        for (int kk = 0; kk < 64; kk += 4) {
            const int kg = kc * 64 + kk + 2 * lhalf;
            v2f b;
            b[0] = W[(kg + 0) * H_ + ntile * 16 + lmod];
            b[1] = W[(kg + 1) * H_ + ntile * 16 + lmod];
            #pragma unroll
            for (int r = 0; r < 4; ++r) {
                v2f a;
                a[0] = xs[cur][r * 16 + lmod][kk + 2 * lhalf + 0];
                a[1] = xs[cur][r * 16 + lmod][kk + 2 * lhalf + 1];
                acc[r] = WMMA_F32(a, b, acc[r]);
            }
        }
        __syncthreads();            // everyone done reading buf `cur`
    }

    // D layout: lane -> col = lmod, row = j + 8*lhalf (per row-tile r)
    #pragma unroll
    for (int r = 0; r < 4; ++r)
        #pragma unroll
        for (int j = 0; j < 8; ++j)
            Out[(rowBase + r * 16 + j + 8 * lhalf) * H_ + ntile * 16 + lmod] = acc[r][j];
}

// ---------------------------------------------------------------------------
// Stage 2: causal flash attention.  Grid (T/64, B), 128 threads (4 waves).
// Wave w owns query rows [qb*64 + w*16, +16).  Key blocks of 64,
// K/V tiles double-buffered in LDS via async copies.
// ---------------------------------------------------------------------------
__global__ __launch_bounds__(128)
void attn_kernel(const float* __restrict__ Q,
                 const float* __restrict__ K,
                 const float* __restrict__ V,
                 float* __restrict__ out)
{
    __shared__ float Ks[2][64][68];
    __shared__ float Vs[2][64][68];
    __shared__ float Ps[64][68];

    const int tid   = threadIdx.x;
    const int wave  = tid >> 5;
    const int lane  = tid & 31;
    const int lhalf = lane >> 4;
    const int lmod  = lane & 15;
    const int qb    = blockIdx.x;     // query block of 64
    const int bt    = blockIdx.y;     // batch
    const long qrowBase = (long)bt * T_ + qb * 64 + wave * 16;
    const float scale = 0.03125f;     // C^-0.5 = 1/32 (reference scales by C, not H)

    // 2 tiles x 1024 float4 / 128 threads -> 16 issues per thread, uniform.
    auto issue_kv = [&](int kb, int buf) {
        const long kBase = (long)bt * T_ + (long)kb * 64;
        #pragma unroll
        for (int it = 0; it < 8; ++it) {
            const int e = tid + it * 128;
            const int r = e >> 4, c = (e & 15) << 2;
            async_copy_b128(&Ks[buf][r][c], &K[(kBase + r) * H_ + c]);
            async_copy_b128(&Vs[buf][r][c], &V[(kBase + r) * H_ + c]);
        }
    };

    // Preload Q A-fragments: 16 k-steps of 4, 2 VGPRs each.
    v2f qa[16];
    #pragma unroll
    for (int kk = 0; kk < 16; ++kk) {
        const int k0 = kk * 4 + 2 * lhalf;
        const float2 t = *(const float2*)&Q[(qrowBase + lmod) * H_ + k0];
        qa[kk][0] = t.x; qa[kk][1] = t.y;
    }

    v8f O[4];
    #pragma unroll
    for (int i = 0; i < 4; ++i) O[i] = (v8f){0,0,0,0,0,0,0,0};
    float mrun[8], lrun[8];
    #pragma unroll
    for (int j = 0; j < 8; ++j) { mrun[j] = -1e30f; lrun[j] = 0.0f; }

    issue_kv(0, 0);
    for (int kb = 0; kb <= qb; ++kb) {
        const int cur = kb & 1;
        if (kb < qb) {
            issue_kv(kb + 1, cur ^ 1);
            WAIT_ASYNC(16);         // 16 next-tile copies may stay in flight
        } else {
            WAIT_ASYNC(0);
        }
        __syncthreads();

        // S = Q K^T for this wave's 16 rows x 64 keys (4 N-tiles)
        const bool diag = (kb == qb);
        v8f st[4];
        #pragma unroll
        for (int i = 0; i < 4; ++i) {
            v8f s = (v8f){0,0,0,0,0,0,0,0};
            #pragma unroll
            for (int kk = 0; kk < 16; ++kk) {
                const int k0 = kk * 4 + 2 * lhalf;
                v2f b;
                b[0] = Ks[cur][i * 16 + lmod][k0 + 0];
                b[1] = Ks[cur][i * 16 + lmod][k0 + 1];
                s = WMMA_F32(qa[kk], b, s);
            }
            const int key = kb * 64 + i * 16 + lmod;
            #pragma unroll
            for (int j = 0; j < 8; ++j) {
                float sv = s[j] * scale;
                const int row = qb * 64 + wave * 16 + j + 8 * lhalf;
                if (diag && key > row) sv = -1e30f;
                s[j] = sv;
            }
            st[i] = s;
        }

        // online softmax: per-lane component j is row (j + 8*lhalf);
        // reduce across the 16 lanes of each half (xor 1,2,4,8 stays in-half)
        float mnew[8], fac[8];
        #pragma unroll
        for (int j = 0; j < 8; ++j) {
            float mx = fmaxf(fmaxf(st[0][j], st[1][j]), fmaxf(st[2][j], st[3][j]));
            #pragma unroll
            for (int off = 1; off < 16; off <<= 1)
                mx = fmaxf(mx, __shfl_xor(mx, off));
            mnew[j] = fmaxf(mrun[j], mx);
            fac[j]  = __expf(mrun[j] - mnew[j]);
            mrun[j] = mnew[j];
        }
        #pragma unroll
        for (int i = 0; i < 4; ++i)
            #pragma unroll
            for (int j = 0; j < 8; ++j)
                st[i][j] = __expf(st[i][j] - mnew[j]);
        #pragma unroll
        for (int j = 0; j < 8; ++j) {
            float sm = st[0][j] + st[1][j] + st[2][j] + st[3][j];
            #pragma unroll
            for (int off = 1; off < 16; off <<= 1)
                sm += __shfl_xor(sm, off);
            lrun[j] = lrun[j] * fac[j] + sm;
        }
        #pragma unroll
        for (int i = 0; i < 4; ++i)
            #pragma unroll
            for (int j = 0; j < 8; ++j)
                O[i][j] *= fac[j];

        // transpose P (D layout) -> LDS so we can re-read it as A-fragments
        #pragma unroll
        for (int i = 0; i < 4; ++i)
            #pragma unroll
            for (int j = 0; j < 8; ++j)
                Ps[wave * 16 + j + 8 * lhalf][i * 16 + lmod] = st[i][j];
        __syncthreads();

        // O += P V
        #pragma unroll
        for (int kk = 0; kk < 16; ++kk) {
            const int k0 = kk * 4 + 2 * lhalf;
            v2f a;
            a[0] = Ps[wave * 16 + lmod][k0 + 0];
            a[1] = Ps[wave * 16 + lmod][k0 + 1];
            #pragma unroll
            for (int i = 0; i < 4; ++i) {
                v2f b;
                b[0] = Vs[cur][k0 + 0][i * 16 + lmod];
                b[1] = Vs[cur][k0 + 1][i * 16 + lmod];
                O[i] = WMMA_F32(a, b, O[i]);
            }
        }
        __syncthreads();            // all reads of buf `cur` and Ps done
    }

    // epilogue: divide by row sums, write [B,T,H]
    #pragma unroll
    for (int i = 0; i < 4; ++i)
        #pragma unroll
        for (int j = 0; j < 8; ++j)
            out[(qrowBase + j + 8 * lhalf) * H_ + i * 16 + lmod] = O[i][j] / lrun[j];
}

// ---------------------------------------------------------------------------
extern "C" void kernel_launch(void* const* d_in, const int* in_sizes, int n_in,
                              void* d_out, int out_size, void* d_ws, size_t ws_size,
                              hipStream_t stream) {
    (void)in_sizes; (void)n_in; (void)out_size; (void)ws_size;
    const float* x  = (const float*)d_in[0];
    const float* Wq = (const float*)d_in[1];
    const float* Wk = (const float*)d_in[2];
    const float* Wv = (const float*)d_in[3];

    float* ws = (float*)d_ws;
    const size_t per = (size_t)B_ * T_ * H_;   // 1M floats each
    float* Q = ws;
    float* K = ws + per;
    float* V = ws + 2 * per;

    qkv_proj_kernel<<<(B_ * T_) / 64, 384, 0, stream>>>(x, Wq, Wk, Wv, Q, K, V);
    attn_kernel<<<dim3(T_ / 64, B_), 128, 0, stream>>>(Q, K, V, (float*)d_out);
}